// EDM_task2_85212151153316
// MI455X (gfx1250) — compile-verified
//
#include <hip/hip_runtime.h>
#include <hip/hip_bf16.h>

// Problem constants (match reference)
#define BB   8
#define MO   12
#define DD   512
#define NO   37
#define NA   6
#define NTR  3
#define EMB  128
#define NAA  100
#define NC   300
#define KK   72      // MO*NA
#define PP   5112    // KK*(KK-1)
#define HALFD 768    // D + 2*EMB

typedef __attribute__((ext_vector_type(2))) float v2f;
typedef __attribute__((ext_vector_type(8))) float v8f;

// fp32 WMMA: D(16x16) += A(16x4) * B(4x16), exact fp32 math.
__device__ __forceinline__ v8f wmma_k4(v2f a, v2f b, v8f c) {
#if defined(__HIP_DEVICE_COMPILE__) && __has_builtin(__builtin_amdgcn_wmma_f32_16x16x4_f32)
  return __builtin_amdgcn_wmma_f32_16x16x4_f32(false, a, false, b, (short)0, c,
                                               false, false);
#else
  (void)a; (void)b;
  return c; // host pass / fallback (never executed on device)
#endif
}

// guaranteed 8-byte vector load (ka is always even -> 8B aligned)
__device__ __forceinline__ v2f ld2(const float* __restrict__ p) {
  return *(const v2f*)p;
}

// ---------------------------------------------------------------------------
// K0: init segment buffers + invalid-row logit constants
// invconst[t] = b_tr[t] - sum_d W_tr[t,d]   (pair_out == -1 rows)
// ---------------------------------------------------------------------------
__global__ void k_init(const float* __restrict__ W_tr, const float* __restrict__ b_tr,
                       float* __restrict__ invc, unsigned* __restrict__ segenc,
                       int* __restrict__ cnts) {
  int idx = blockIdx.x * blockDim.x + threadIdx.x;
  if (idx < BB * NC) { segenc[idx] = 0u; cnts[idx] = 0; }
  if (idx < NTR) {
    float s = 0.f;
    for (int d = 0; d < 2 * HALFD; ++d) s += W_tr[idx * 2 * HALFD + d];
    invc[idx] = b_tr[idx] - s;
  }
}

// ---------------------------------------------------------------------------
// K1: obj_out / act_out fused GEMM.  A = inp [96,512], B^T = [W_obj;W_act] [43,512]
// One wave per 16x16 tile; mtiles=6, ntiles=3 (48 cols, 43 used).
// Out-of-range columns load a dummy valid row (garbage stays in unwritten cols).
// ---------------------------------------------------------------------------
__global__ void k_objact_gemm(const float* __restrict__ inp,
                              const float* __restrict__ W_obj, const float* __restrict__ b_obj,
                              const float* __restrict__ W_act, const float* __restrict__ b_act,
                              float* __restrict__ out_act, float* __restrict__ out_obj) {
  const int lane = threadIdx.x;
  const int mtile = blockIdx.x / 3, ntile = blockIdx.x % 3;
  const int g = lane >> 4, l = lane & 15;
  const int m = mtile * 16 + l;                 // < 96 always
  const int n0 = ntile * 16 + l;
  const int n = (n0 < NO + NA) ? n0 : 0;        // clamp: garbage -> unwritten cols
  const float* Arow = inp + m * DD;
  const float* Brow = (n < NO) ? (W_obj + n * DD) : (W_act + (n - NO) * DD);
  v8f c = {};
#pragma unroll 8
  for (int k0 = 0; k0 < DD; k0 += 4) {
    const int ka = k0 + 2 * g;
    c = wmma_k4(ld2(Arow + ka), ld2(Brow + ka), c);
  }
  const int col = ntile * 16 + l;
  for (int r = 0; r < 8; ++r) {
    const int row = mtile * 16 + 8 * g + r;     // (b*MO + o)
    const float v = c[r];
    if (col < NO)            out_obj[row * NO + col] = v + b_obj[col];
    else if (col < NO + NA)  out_act[row * NA + (col - NO)] = v + b_act[col - NO];
  }
}

// ---------------------------------------------------------------------------
// K1b: masked mean pooling  pooled[b,d]
// ---------------------------------------------------------------------------
__global__ void k_pool(const float* __restrict__ inp, const float* __restrict__ objmask,
                       float* __restrict__ pooled) {
  int idx = blockIdx.x * blockDim.x + threadIdx.x;
  if (idx >= BB * DD) return;
  const int b = idx >> 9, d = idx & (DD - 1);
  float s = 0.f, dn = 0.f;
  for (int o = 0; o < MO; ++o) {
    const float mk = objmask[b * MO + o];
    s += inp[(b * MO + o) * DD + d] * mk;
    dn += mk;
  }
  pooled[idx] = s / dn;
}

// ---------------------------------------------------------------------------
// K2: non_exist_out = pooled @ W_ne^T + b_ne.  M=8 (pad 16), N=300 (19 tiles), K=512
// Out-of-range rows/cols clamp to row 0 (garbage stays in unwritten outputs).
// ---------------------------------------------------------------------------
__global__ void k_ne_gemm(const float* __restrict__ pooled, const float* __restrict__ W_ne,
                          const float* __restrict__ b_ne, float* __restrict__ nonex) {
  const int lane = threadIdx.x;
  const int ntile = blockIdx.x;
  const int g = lane >> 4, l = lane & 15;
  const int m = (l < BB) ? l : 0;               // clamp
  const int n0 = ntile * 16 + l;
  const int n = (n0 < NC) ? n0 : 0;             // clamp
  const float* Arow = pooled + m * DD;
  const float* Brow = W_ne + n * DD;
  v8f c = {};
#pragma unroll 8
  for (int k0 = 0; k0 < DD; k0 += 4) {
    const int ka = k0 + 2 * g;
    c = wmma_k4(ld2(Arow + ka), ld2(Brow + ka), c);
  }
  const int col = ntile * 16 + l;
  if (col < NC) {
    for (int r = 0; r < 8; ++r) {
      const int row = 8 * g + r;
      if (row < BB) nonex[row * NC + col] = c[r] + b_ne[col];
    }
  }
}

// ---------------------------------------------------------------------------
// K3: per (b,k) slot: selected mask, aaidx, argmax object class
// ---------------------------------------------------------------------------
__global__ void k_classify(const float* __restrict__ objmask, const int* __restrict__ AAidxs,
                           const float* __restrict__ act_out, const float* __restrict__ obj_out,
                           int* __restrict__ selM, int* __restrict__ aaidx,
                           int* __restrict__ pobj) {
  int idx = blockIdx.x * blockDim.x + threadIdx.x;
  if (idx >= BB * KK) return;
  const int b = idx / KK, k = idx % KK;
  const int o = k / NA, a = k % NA;
  const float logit = act_out[(b * MO + o) * NA + a];
  const float mk = objmask[b * MO + o];
  const int tgt = AAidxs[idx];
  // sigmoid(x) > 0.5  <=>  x > 0 (strict)
  const int pred = (logit > 0.0f) ? 1 : 0;
  const int sel = (pred && (mk == 1.0f) && (tgt != -1)) ? 1 : 0;
  selM[idx] = sel;
  aaidx[idx] = sel ? tgt : -1;
  const float* row = obj_out + (b * MO + o) * NO;
  int best = 0; float bv = row[0];
  for (int c2 = 1; c2 < NO; ++c2) { const float v = row[c2]; if (v > bv) { bv = v; best = c2; } }
  pobj[idx] = best;
}

// ---------------------------------------------------------------------------
// K3b: per-batch sorted selected-index list (exactly the stable-argsort packing)
// ---------------------------------------------------------------------------
__global__ void k_scan(const int* __restrict__ selM, int* __restrict__ Slist,
                       int* __restrict__ Scnt) {
  const int b = threadIdx.x;
  if (b >= BB) return;
  int s = 0;
  for (int k = 0; k < KK; ++k)
    if (selM[b * KK + k]) Slist[b * KK + (s++)] = k;
  Scnt[b] = s;
}

// ---------------------------------------------------------------------------
// K4: halfAB[row,t] = dot(comp[row], Wcat[t]) ; comp gathered on the fly.
// Wcat rows 0..2 = W_tr[:, :768], 3..5 = W_tr[:, 768:].  M=576, N=16, K=768.
// K-loop split into three branchless segments: inp | obj_emb | act_emb.
// ---------------------------------------------------------------------------
__global__ void k_half_gemm(const float* __restrict__ inp, const float* __restrict__ obj_emb,
                            const float* __restrict__ act_emb, const float* __restrict__ W_tr,
                            const int* __restrict__ pobj, float* __restrict__ halfAB) {
  const int lane = threadIdx.x;
  const int mtile = blockIdx.x;
  const int g = lane >> 4, l = lane & 15;
  const int row = mtile * 16 + l;     // < 576
  const int b = row / KK, k = row % KK;
  const int o = k / NA, a = k % NA;
  const float* inprow = inp + (b * MO + o) * DD;
  const float* oerow  = obj_emb + pobj[row] * EMB;
  const float* aerow  = act_emb + a * EMB;
  const int t = (l < 2 * NTR) ? l : 0;          // clamp: garbage -> unwritten cols
  const float* Bt = (t < NTR) ? (W_tr + t * 2 * HALFD)
                              : (W_tr + (t - NTR) * 2 * HALFD + HALFD);
  v8f c = {};
#pragma unroll 8
  for (int k0 = 0; k0 < DD; k0 += 4) {          // comp[0:512) = inp row
    const int ka = k0 + 2 * g;
    c = wmma_k4(ld2(inprow + ka), ld2(Bt + ka), c);
  }
  const float* Bt1 = Bt + DD;
#pragma unroll 8
  for (int k0 = 0; k0 < EMB; k0 += 4) {         // comp[512:640) = obj_emb row
    const int ka = k0 + 2 * g;
    c = wmma_k4(ld2(oerow + ka), ld2(Bt1 + ka), c);
  }
  const float* Bt2 = Bt + DD + EMB;
#pragma unroll 8
  for (int k0 = 0; k0 < EMB; k0 += 4) {         // comp[640:768) = act_emb row
    const int ka = k0 + 2 * g;
    c = wmma_k4(ld2(aerow + ka), ld2(Bt2 + ka), c);
  }
  const int col = l;
  if (col < 2 * NTR) {
    for (int r = 0; r < 8; ++r) {
      const int orow = mtile * 16 + 8 * g + r;
      halfAB[orow * (2 * NTR) + col] = c[r];
    }
  }
}

// monotonic float->uint encoding so unsigned atomicMax == float max
__device__ __forceinline__ unsigned enc_f(float f) {
  unsigned b = __float_as_uint(f);
  return (b & 0x80000000u) ? ~b : (b | 0x80000000u);
}
__device__ __forceinline__ float dec_f(unsigned e) {
  unsigned b = (e & 0x80000000u) ? (e ^ 0x80000000u) : ~e;
  return __uint_as_float(b);
}

// ---------------------------------------------------------------------------
// K5: packed pair assembly + logits + targets + class-wise segment max
// ---------------------------------------------------------------------------
__global__ void k_pairs(const float* __restrict__ halfAB, const float* __restrict__ b_tr,
                        const float* __restrict__ invc, const int* __restrict__ Slist,
                        const int* __restrict__ Scnt, const int* __restrict__ aaidx,
                        const int* __restrict__ lookup, const float* __restrict__ TR_tgts,
                        float* __restrict__ pair_logits, float* __restrict__ tr_targets,
                        unsigned* __restrict__ segenc, int* __restrict__ cnts) {
  int idx = blockIdx.x * blockDim.x + threadIdx.x;
  if (idx >= BB * PP) return;
  const int b = idx / PP, q = idx % PP;
  const int s = Scnt[b];
  const int nv = s * (s - 1);
  float* pl = pair_logits + idx * NTR;
  float* tt = tr_targets + idx * NTR;
  if (q < nv) {
    const int a = q / (s - 1), r = q % (s - 1);
    const int i = Slist[b * KK + a];
    const int j = Slist[b * KK + r + (r >= a ? 1 : 0)];
    const int ai = aaidx[b * KK + i];
    const int aj = aaidx[b * KK + j];
    const float* hi = halfAB + (b * KK + i) * (2 * NTR);
    const float* hj = halfAB + (b * KK + j) * (2 * NTR);
    const float* tr = TR_tgts + ((b * KK + i) * KK + j) * NTR;
    const int* lk = lookup + (ai * NAA + aj) * NTR;
    for (int t = 0; t < NTR; ++t) {
      const float L = hi[t] + hj[NTR + t] + b_tr[t];
      pl[t] = L;
      tt[t] = tr[t];
      const int c2 = lk[t];
      atomicMax(&segenc[b * NC + c2], enc_f(L));
      atomicAdd(&cnts[b * NC + c2], 1);
    }
  } else {
    for (int t = 0; t < NTR; ++t) { pl[t] = invc[t]; tt[t] = -1.0f; }
  }
}

// ---------------------------------------------------------------------------
// K6: final merge
// ---------------------------------------------------------------------------
__global__ void k_final(const unsigned* __restrict__ segenc, const int* __restrict__ cnts,
                        const float* __restrict__ nonex, float* __restrict__ outf) {
  int idx = blockIdx.x * blockDim.x + threadIdx.x;
  if (idx >= BB * NC) return;
  outf[idx] = (cnts[idx] > 0) ? dec_f(segenc[idx]) : nonex[idx];
}

// ---------------------------------------------------------------------------
extern "C" void kernel_launch(void* const* d_in, const int* in_sizes, int n_in,
                              void* d_out, int out_size, void* d_ws, size_t ws_size,
                              hipStream_t stream) {
  (void)in_sizes; (void)n_in; (void)out_size; (void)ws_size;
  const float* inp      = (const float*)d_in[0];
  const float* objmask  = (const float*)d_in[1];
  const float* TR_tgts  = (const float*)d_in[2];
  const float* W_obj    = (const float*)d_in[3];
  const float* b_obj    = (const float*)d_in[4];
  const float* W_act    = (const float*)d_in[5];
  const float* b_act    = (const float*)d_in[6];
  const float* W_tr     = (const float*)d_in[7];
  const float* b_tr     = (const float*)d_in[8];
  const float* W_ne     = (const float*)d_in[9];
  const float* b_ne     = (const float*)d_in[10];
  const float* obj_emb  = (const float*)d_in[11];
  const float* act_emb  = (const float*)d_in[12];
  const int*   AAidxs   = (const int*)d_in[13];
  // d_in[14] = obj_act_lookup: eval-branch only, unused
  const int*   aalookup = (const int*)d_in[15];

  // output layout (reference tuple order, flat)
  float* out_final  = (float*)d_out;                 // [8,300]
  float* out_plog   = out_final + BB * NC;           // [8,5112,3]
  float* out_trt    = out_plog + BB * PP * NTR;      // [8,5112,3]
  float* out_act    = out_trt + BB * PP * NTR;       // [8,12,6]
  float* out_obj    = out_act + BB * MO * NA;        // [8,12,37]

  // workspace layout
  float* pooled  = (float*)d_ws;                     // 4096
  float* nonex   = pooled + BB * DD;                 // 2400
  float* halfAB  = nonex + BB * NC;                  // 3456
  float* invc    = halfAB + BB * KK * 2 * NTR;       // 4 (padded)
  int*   selM    = (int*)(invc + 4);                 // 576
  int*   aaidx   = selM + BB * KK;                   // 576
  int*   pobj    = aaidx + BB * KK;                  // 576
  int*   Slist   = pobj + BB * KK;                   // 576
  int*   Scnt    = Slist + BB * KK;                  // 8
  unsigned* segenc = (unsigned*)(Scnt + BB);         // 2400
  int*   cnts    = (int*)(segenc + BB * NC);         // 2400

  k_init<<<(BB * NC + 255) / 256, 256, 0, stream>>>(W_tr, b_tr, invc, segenc, cnts);
  k_objact_gemm<<<6 * 3, 32, 0, stream>>>(inp, W_obj, b_obj, W_act, b_act, out_act, out_obj);
  k_pool<<<(BB * DD + 255) / 256, 256, 0, stream>>>(inp, objmask, pooled);
  k_ne_gemm<<<(NC + 15) / 16, 32, 0, stream>>>(pooled, W_ne, b_ne, nonex);
  k_classify<<<(BB * KK + 255) / 256, 256, 0, stream>>>(objmask, AAidxs, out_act, out_obj,
                                                        selM, aaidx, pobj);
  k_scan<<<1, BB, 0, stream>>>(selM, Slist, Scnt);
  k_half_gemm<<<(BB * KK) / 16, 32, 0, stream>>>(inp, obj_emb, act_emb, W_tr, pobj, halfAB);
  k_pairs<<<(BB * PP + 255) / 256, 256, 0, stream>>>(halfAB, b_tr, invc, Slist, Scnt, aaidx,
                                                     aalookup, TR_tgts, out_plog, out_trt,
                                                     segenc, cnts);
  k_final<<<(BB * NC + 255) / 256, 256, 0, stream>>>(segenc, cnts, nonex, out_final);
}